// GNN_23656679867765
// MI455X (gfx1250) — compile-verified
//
#include <hip/hip_runtime.h>

// CDNA5 / gfx1250. wave32. Full-fp32 WMMA (V_WMMA_F32_16X16X4_F32).
typedef float v2f __attribute__((ext_vector_type(2)));
typedef float v8f __attribute__((ext_vector_type(8)));

#define TPB 256

// ---------------- init: deg=1 (self loop), zero pool accumulators ----------------
__global__ void gnn_init(float* __restrict__ deg, float* __restrict__ sums,
                         float* __restrict__ cnts, int N, int G) {
  int i = blockIdx.x * blockDim.x + threadIdx.x;
  if (i < N) deg[i] = 1.0f;              // self-loop contributes 1 to in-degree
  if (i < G * 32) sums[i] = 0.0f;
  if (i < G) cnts[i] = 0.0f;
}

// in-degree over dst edges
__global__ void gnn_deg(const int* __restrict__ dst, float* __restrict__ deg, int E) {
  int e = blockIdx.x * blockDim.x + threadIdx.x;
  if (e < E) atomicAdd(&deg[dst[e]], 1.0f);
}

// deg -> dinv in place (deg >= 1 always, self loops)
__global__ void gnn_rsqrt(float* __restrict__ deg, int N) {
  int i = blockIdx.x * blockDim.x + threadIdx.x;
  if (i < N) deg[i] = rsqrtf(deg[i]);
}

// ---------------- layer 1: aggregate FIRST on 13 channels (cheap), then GEMM ----
// self-loop term: agg1[i] = dinv[i]^2 * x[i]
__global__ void gnn_agg1_init(const float* __restrict__ x, const float* __restrict__ dinv,
                              float* __restrict__ agg, int N) {
  int i = blockIdx.x * blockDim.x + threadIdx.x;
  if (i < N) {
    float nrm = dinv[i] * dinv[i];
    #pragma unroll
    for (int c = 0; c < 13; ++c) agg[i * 13 + c] = nrm * x[i * 13 + c];
  }
}

__global__ void gnn_agg1_edge(const float* __restrict__ x, const float* __restrict__ dinv,
                              const int* __restrict__ src, const int* __restrict__ dst,
                              float* __restrict__ agg, int E) {
  int e = blockIdx.x * blockDim.x + threadIdx.x;
  if (e < E) {
    int s = src[e], d = dst[e];
    float nrm = dinv[s] * dinv[d];
    #pragma unroll
    for (int c = 0; c < 13; ++c)
      atomicAdd(&agg[d * 13 + c], nrm * x[s * 13 + c]);
  }
}

// H1 = relu(agg1[N,13] @ W1[13,64] + b1)   -- fp32 WMMA 16x16x4, K padded 13->16.
// 256 threads = 8 waves; each wave computes a 16x64 output tile.
// A 16x4 f32 layout: lanes0-15 M=0..15 {reg0:K=k0, reg1:K=k0+1}; lanes16-31 {K=k0+2, K=k0+3}.
// B 4x16 f32 layout (row striped across lanes per VGPR): reg0 rows {k0 | k0+2}, reg1 {k0+1 | k0+3}.
__global__ __launch_bounds__(256)
void gnn_gemm1_wmma(const float* __restrict__ A, const float* __restrict__ W,
                    const float* __restrict__ bias, float* __restrict__ H, int N) {
  const int lane = threadIdx.x & 31;
  const int wave = threadIdx.x >> 5;
  const int mbase = (blockIdx.x * 8 + wave) * 16;
  const int half = lane >> 4;
  const int l16 = lane & 15;

  v8f acc[4];
  const v8f zero = {0.f, 0.f, 0.f, 0.f, 0.f, 0.f, 0.f, 0.f};
  #pragma unroll
  for (int nt = 0; nt < 4; ++nt) acc[nt] = zero;

  const int rowA = mbase + l16;

  if (mbase + 16 <= N) {
    // ------ full-tile fast path: no row guards, no exec divergence ------
    const float* Arow = A + rowA * 13;
    #pragma unroll
    for (int kk = 0; kk < 12; kk += 4) {
      int ka0 = kk + half * 2;          // <= 10, always valid
      v2f av; av[0] = Arow[ka0]; av[1] = Arow[ka0 + 1];
      #pragma unroll
      for (int nt = 0; nt < 4; ++nt) {
        int col = nt * 16 + l16;
        v2f bv; bv[0] = W[ka0 * 64 + col]; bv[1] = W[(ka0 + 1) * 64 + col];
        acc[nt] = __builtin_amdgcn_wmma_f32_16x16x4_f32(
            false, av, false, bv, (short)0, acc[nt], false, false);
      }
    }
    // last K-step (kk=12): K=13,14,15 are zero padding.
    // half==0 -> reg0=K12 (valid), reg1=K13 (0); half==1 -> reg0=K14 (0), reg1=K15 (0)
    {
      float a12 = Arow[12];
      v2f av; av[0] = half ? 0.0f : a12; av[1] = 0.0f;
      #pragma unroll
      for (int nt = 0; nt < 4; ++nt) {
        int col = nt * 16 + l16;
        float b12 = W[12 * 64 + col];
        v2f bv; bv[0] = half ? 0.0f : b12; bv[1] = 0.0f;
        acc[nt] = __builtin_amdgcn_wmma_f32_16x16x4_f32(
            false, av, false, bv, (short)0, acc[nt], false, false);
      }
    }
    #pragma unroll
    for (int nt = 0; nt < 4; ++nt) {
      int col = nt * 16 + l16;
      float bc = bias[col];
      #pragma unroll
      for (int v = 0; v < 8; ++v) {
        int row = mbase + v + half * 8;
        H[row * 64 + col] = fmaxf(acc[nt][v] + bc, 0.0f);
      }
    }
  } else {
    // ------ tail path (unused when N % 16 == 0) ------
    const bool rv = rowA < N;
    #pragma unroll
    for (int kk = 0; kk < 16; kk += 4) {
      int ka0 = kk + half * 2;
      int ka1 = ka0 + 1;
      float a0 = (rv && ka0 < 13) ? A[rowA * 13 + ka0] : 0.0f;
      float a1 = (rv && ka1 < 13) ? A[rowA * 13 + ka1] : 0.0f;
      v2f av; av[0] = a0; av[1] = a1;
      #pragma unroll
      for (int nt = 0; nt < 4; ++nt) {
        int col = nt * 16 + l16;
        float b0 = (ka0 < 13) ? W[ka0 * 64 + col] : 0.0f;
        float b1 = (ka1 < 13) ? W[ka1 * 64 + col] : 0.0f;
        v2f bv; bv[0] = b0; bv[1] = b1;
        acc[nt] = __builtin_amdgcn_wmma_f32_16x16x4_f32(
            false, av, false, bv, (short)0, acc[nt], false, false);
      }
    }
    #pragma unroll
    for (int nt = 0; nt < 4; ++nt) {
      int col = nt * 16 + l16;
      float bc = bias[col];
      #pragma unroll
      for (int v = 0; v < 8; ++v) {
        int row = mbase + v + half * 8;
        if (row < N) H[row * 64 + col] = fmaxf(acc[nt][v] + bc, 0.0f);
      }
    }
  }
}

// XW2 = H1[N,64] @ W2[64,32]   -- fp32 WMMA, K=64, per-wave 16x32 tile.
__global__ __launch_bounds__(256)
void gnn_gemm2_wmma(const float* __restrict__ A, const float* __restrict__ W,
                    float* __restrict__ O, int N) {
  const int lane = threadIdx.x & 31;
  const int wave = threadIdx.x >> 5;
  const int mbase = (blockIdx.x * 8 + wave) * 16;
  const int half = lane >> 4;
  const int l16 = lane & 15;

  v8f acc[2];
  const v8f zero = {0.f, 0.f, 0.f, 0.f, 0.f, 0.f, 0.f, 0.f};
  acc[0] = zero; acc[1] = zero;

  const int rowA = mbase + l16;

  if (mbase + 16 <= N) {
    // ------ full-tile fast path: unguarded, 8B-aligned float2 A loads ------
    const float* Arow = A + rowA * 64;
    #pragma unroll
    for (int kk = 0; kk < 64; kk += 4) {
      int ka0 = kk + half * 2;                        // even -> 8B aligned
      const float2 a2 = *(const float2*)(Arow + ka0); // global_load_b64
      v2f av; av[0] = a2.x; av[1] = a2.y;
      #pragma unroll
      for (int nt = 0; nt < 2; ++nt) {
        int col = nt * 16 + l16;
        v2f bv; bv[0] = W[ka0 * 32 + col]; bv[1] = W[(ka0 + 1) * 32 + col];
        acc[nt] = __builtin_amdgcn_wmma_f32_16x16x4_f32(
            false, av, false, bv, (short)0, acc[nt], false, false);
      }
    }
    #pragma unroll
    for (int nt = 0; nt < 2; ++nt) {
      int col = nt * 16 + l16;
      #pragma unroll
      for (int v = 0; v < 8; ++v) {
        int row = mbase + v + half * 8;
        O[row * 32 + col] = acc[nt][v];
      }
    }
  } else {
    // ------ tail path (unused when N % 16 == 0) ------
    const bool rv = rowA < N;
    #pragma unroll
    for (int kk = 0; kk < 64; kk += 4) {
      int ka0 = kk + half * 2;
      int ka1 = ka0 + 1;
      float a0 = rv ? A[rowA * 64 + ka0] : 0.0f;
      float a1 = rv ? A[rowA * 64 + ka1] : 0.0f;
      v2f av; av[0] = a0; av[1] = a1;
      #pragma unroll
      for (int nt = 0; nt < 2; ++nt) {
        int col = nt * 16 + l16;
        v2f bv; bv[0] = W[ka0 * 32 + col]; bv[1] = W[ka1 * 32 + col];
        acc[nt] = __builtin_amdgcn_wmma_f32_16x16x4_f32(
            false, av, false, bv, (short)0, acc[nt], false, false);
      }
    }
    #pragma unroll
    for (int nt = 0; nt < 2; ++nt) {
      int col = nt * 16 + l16;
      #pragma unroll
      for (int v = 0; v < 8; ++v) {
        int row = mbase + v + half * 8;
        if (row < N) O[row * 32 + col] = acc[nt][v];
      }
    }
  }
}

// ---------------- layer 2: aggregate AFTER transform (32 channels) ----------------
// self loop: h2[i] = dinv[i]^2 * xw2[i]   (float4 vectorized: 8 float4 per row)
__global__ void gnn_h2_init(const float* __restrict__ xw2, const float* __restrict__ dinv,
                            float* __restrict__ h2, int N) {
  int t = blockIdx.x * blockDim.x + threadIdx.x;
  if (t < N * 8) {
    int i = t >> 3;
    float nrm = dinv[i] * dinv[i];
    float4 v = ((const float4*)xw2)[t];
    float4 o; o.x = v.x * nrm; o.y = v.y * nrm; o.z = v.z * nrm; o.w = v.w * nrm;
    ((float4*)h2)[t] = o;
  }
}

// 4 threads per edge, 8 channels each
__global__ void gnn_h2_edge(const float* __restrict__ xw2, const float* __restrict__ dinv,
                            const int* __restrict__ src, const int* __restrict__ dst,
                            float* __restrict__ h2, int E) {
  int t = blockIdx.x * blockDim.x + threadIdx.x;
  if (t < E * 4) {
    int e = t >> 2, q = (t & 3) * 8;
    int s = src[e], d = dst[e];
    float nrm = dinv[s] * dinv[d];
    #pragma unroll
    for (int c = 0; c < 8; ++c)
      atomicAdd(&h2[d * 32 + q + c], nrm * xw2[s * 32 + q + c]);
  }
}

// ---------------- global mean pool ----------------
// batch is sorted: each thread walks 8 consecutive nodes, accumulates per-segment
// in registers, flushes atomics only on batch-id change (~8x fewer atomics).
__global__ void gnn_pool(const float* __restrict__ h2, const int* __restrict__ batch,
                         float* __restrict__ sums, float* __restrict__ cnts, int N) {
  int t = blockIdx.x * blockDim.x + threadIdx.x;
  int nblk = (N + 7) >> 3;
  if (t >= nblk * 4) return;
  int blk = t >> 2, q = (t & 3) * 8;
  int i0 = blk * 8;
  int iend = (i0 + 8 < N) ? (i0 + 8) : N;
  int cur = batch[i0];
  float acc[8];
  #pragma unroll
  for (int c = 0; c < 8; ++c) acc[c] = 0.0f;
  float cnt = 0.0f;
  for (int i = i0; i < iend; ++i) {
    int b = batch[i];
    if (b != cur) {
      #pragma unroll
      for (int c = 0; c < 8; ++c) atomicAdd(&sums[cur * 32 + q + c], acc[c]);
      if (q == 0) atomicAdd(&cnts[cur], cnt);
      #pragma unroll
      for (int c = 0; c < 8; ++c) acc[c] = 0.0f;
      cnt = 0.0f;
      cur = b;
    }
    #pragma unroll
    for (int c = 0; c < 8; ++c) acc[c] += h2[i * 32 + q + c];
    cnt += 1.0f;
  }
  #pragma unroll
  for (int c = 0; c < 8; ++c) atomicAdd(&sums[cur * 32 + q + c], acc[c]);
  if (q == 0) atomicAdd(&cnts[cur], cnt);
}

// g = sums / max(cnt,1) + b2   (b2 folded here: mean(h + b2) = mean(h) + b2)
__global__ void gnn_gmean(const float* __restrict__ sums, const float* __restrict__ cnts,
                          const float* __restrict__ b2, float* __restrict__ g, int G) {
  int t = blockIdx.x * blockDim.x + threadIdx.x;
  if (t < G * 32) {
    int b = t >> 5, j = t & 31;
    g[t] = sums[t] / fmaxf(cnts[b], 1.0f) + b2[j];
  }
}

// ---------------- FC head (tiny: 512x32, plain VALU) ----------------
__global__ void gnn_fc_relu(const float* __restrict__ g, const float* __restrict__ w,
                            const float* __restrict__ bias, float* __restrict__ o, int G) {
  int t = blockIdx.x * blockDim.x + threadIdx.x;
  if (t < G * 32) {
    int b = t >> 5, j = t & 31;
    float s = bias[j];
    #pragma unroll
    for (int k = 0; k < 32; ++k) s += g[b * 32 + k] * w[k * 32 + j];
    o[t] = fmaxf(s, 0.0f);
  }
}

__global__ void gnn_fc_out(const float* __restrict__ g, const float* __restrict__ w,
                           const float* __restrict__ bias, float* __restrict__ o, int G) {
  int t = blockIdx.x * blockDim.x + threadIdx.x;
  if (t < G * 32) {
    int b = t >> 5, j = t & 31;
    float s = bias[j];
    #pragma unroll
    for (int k = 0; k < 32; ++k) s += g[b * 32 + k] * w[k * 32 + j];
    o[t] = s;
  }
}

extern "C" void kernel_launch(void* const* d_in, const int* in_sizes, int n_in,
                              void* d_out, int out_size, void* d_ws, size_t ws_size,
                              hipStream_t stream) {
  const float* x    = (const float*)d_in[0];
  const int*   ei   = (const int*)d_in[1];
  const int*   batch= (const int*)d_in[2];
  const float* W1   = (const float*)d_in[3];
  const float* b1   = (const float*)d_in[4];
  const float* W2   = (const float*)d_in[5];
  const float* b2   = (const float*)d_in[6];
  const float* fc1w = (const float*)d_in[7];
  const float* fc1b = (const float*)d_in[8];
  const float* fc2w = (const float*)d_in[9];
  const float* fc2b = (const float*)d_in[10];

  const int N = in_sizes[0] / 13;
  const int E = in_sizes[1] / 2;
  const int G = out_size / 32;
  const int* src = ei;
  const int* dst = ei + E;

  // workspace carve-up (float counts, each region 16B-aligned)
  auto al4 = [](size_t v) { return (v + 3) & ~(size_t)3; };
  float* ws   = (float*)d_ws;
  size_t off  = 0;
  float* dinv = ws + off; off = al4(off + (size_t)N);          // deg, then dinv in place
  float* agg1 = ws + off; off = al4(off + (size_t)13 * N);
  float* h1   = ws + off; off = al4(off + (size_t)64 * N);
  float* xw2  = ws + off; off = al4(off + (size_t)32 * N);
  float* h2   = ws + off; off = al4(off + (size_t)32 * N);
  float* sums = ws + off; off = al4(off + (size_t)32 * G);
  float* cnts = ws + off; off = al4(off + (size_t)G);
  float* g    = ws + off; off = al4(off + (size_t)32 * G);
  float* hfc  = ws + off; off = al4(off + (size_t)32 * G);

  auto cdiv = [](int a, int b) { return (a + b - 1) / b; };

  // degree + normalization
  gnn_init<<<cdiv(N, TPB), TPB, 0, stream>>>(dinv, sums, cnts, N, G);
  gnn_deg<<<cdiv(E, TPB), TPB, 0, stream>>>(dst, dinv, E);
  gnn_rsqrt<<<cdiv(N, TPB), TPB, 0, stream>>>(dinv, N);

  // layer 1: aggregate (13 ch) then WMMA transform 13->64 (+bias+relu)
  gnn_agg1_init<<<cdiv(N, TPB), TPB, 0, stream>>>(x, dinv, agg1, N);
  gnn_agg1_edge<<<cdiv(E, TPB), TPB, 0, stream>>>(x, dinv, src, dst, agg1, E);
  int mt = cdiv(N, 16);
  gnn_gemm1_wmma<<<cdiv(mt, 8), 256, 0, stream>>>(agg1, W1, b1, h1, N);

  // layer 2: WMMA transform 64->32 first, then aggregate (32 ch)
  gnn_gemm2_wmma<<<cdiv(mt, 8), 256, 0, stream>>>(h1, W2, xw2, N);
  gnn_h2_init<<<cdiv(N * 8, TPB), TPB, 0, stream>>>(xw2, dinv, h2, N);
  gnn_h2_edge<<<cdiv(E * 4, TPB), TPB, 0, stream>>>(xw2, dinv, src, dst, h2, E);

  // mean pool (b2 folded into mean) + FC head
  int nblk = (N + 7) / 8;
  gnn_pool<<<cdiv(nblk * 4, TPB), TPB, 0, stream>>>(h2, batch, sums, cnts, N);
  gnn_gmean<<<cdiv(G * 32, TPB), TPB, 0, stream>>>(sums, cnts, b2, g, G);
  gnn_fc_relu<<<cdiv(G * 32, TPB), TPB, 0, stream>>>(g, fc1w, fc1b, hfc, G);
  gnn_fc_out<<<cdiv(G * 32, TPB), TPB, 0, stream>>>(hfc, fc2w, fc2b, (float*)d_out, G);
}